// OnlinePhraseBuilder_8615704396086
// MI455X (gfx1250) — compile-verified
//
#include <hip/hip_runtime.h>
#include <hip/hip_bf16.h>
#include <stdint.h>

// Problem constants (match reference)
#define BB    64
#define TT    8192
#define NW    256        // 32-bit words per row (TT/32)
#define MAXP  32
#define PADID 0

// Workspace layout per row, in int32 units
#define WS_WORDS  0       // 256 words: is_end bitmask
#define WS_PREFIX 256     // 256 words: exclusive phrase count before word w
#define WS_META   512     // [0]=nb (boundary count), [1]=last_real
#define WS_BND    528     // up to 8192 boundary positions
#define WS_ROW    8736    // per-row stride (ints), 16B aligned

// Output layout (floats, concatenated in reference return order)
#define OUT_MASK  ((size_t)0)
#define OUT_TIDX  ((size_t)BB * TT * MAXP)            // 16,777,216
#define OUT_ENDP  ((size_t)2 * BB * TT * MAXP)        // 33,554,432
#define OUT_PID   (OUT_ENDP + (size_t)BB * TT)        // 34,078,720

// ---------------------------------------------------------------------------
// K1: background fill. mask+token_idx regions -> 0.0f, end_pos region -> -1.0f
// (phrase_id region is fully overwritten by K3). Pure b128 store stream.
// ---------------------------------------------------------------------------
__global__ void opb_fill_kernel(float* __restrict__ out) {
    const size_t n4 = OUT_PID / 4;                 // total float4s to fill
    size_t i = (size_t)blockIdx.x * blockDim.x + threadIdx.x;
    const size_t stride = (size_t)gridDim.x * blockDim.x;
    for (; i < n4; i += stride) {
        size_t f = i * 4;
        float v = (f >= OUT_ENDP) ? -1.0f : 0.0f;  // region boundary is /4 aligned
        float4 q = make_float4(v, v, v, v);
        *reinterpret_cast<float4*>(out + f) = q;   // global_store_b128
    }
}

// ---------------------------------------------------------------------------
// K2: per-row scan. One 256-thread block per row.
//  - Stage the row of input_ids into LDS via gfx1250 async-to-LDS (bypasses
//    VGPRs), wait with s_wait_asynccnt.
//  - 256 lanes build base_end bitmask (1 word / lane) + last_real reduction.
//  - Lane 0 walks the boundary chain: next = min(firstE(lb+1), lb+32, last_real)
//    and emits: is_end words, per-word phrase-count prefix, boundary list.
// ---------------------------------------------------------------------------
__global__ void opb_scan_kernel(const int* __restrict__ ids,
                                const int* __restrict__ punct,
                                const int* __restrict__ abbr,
                                int* __restrict__ ws) {
    __shared__ int      lds_ids[TT];     // 32 KB
    __shared__ unsigned e_words[NW];     // base_end bitmask
    __shared__ unsigned is_words[NW];    // is_end bitmask
    __shared__ int      s_last_real;

    const int b   = blockIdx.x;
    const int tid = threadIdx.x;
    const int* row = ids + (size_t)b * TT;

    if (tid == 0) s_last_real = -1;

    // ---- async copy row -> LDS: 8 iters x 256 lanes x 16B = 32 KB ----
    {
        unsigned lds_base = (unsigned)(uintptr_t)(&lds_ids[0]); // LDS byte addr
        #pragma unroll
        for (int it = 0; it < 8; ++it) {
            unsigned byteoff = (unsigned)((it * 256 + tid) * 16);
            unsigned ldsaddr = lds_base + byteoff;
            asm volatile("global_load_async_to_lds_b128 %0, %1, %2 offset:0"
                         :: "v"(ldsaddr), "v"(byteoff), "s"(row)
                         : "memory");
        }
        asm volatile("s_wait_asynccnt 0x0" ::: "memory");
    }
    __syncthreads();   // also orders the s_last_real init vs. atomics below

    // ---- build base_end word for tokens [tid*32, tid*32+32) ----
    {
        const int w = tid;
        unsigned ew = 0u;
        int localmax = -1;
        #pragma unroll 4
        for (int i = 0; i < 32; ++i) {
            int t  = (w << 5) | i;
            int id = lds_ids[t];
            if (id != PADID) localmax = t;
            int prev = (t == 0) ? PADID : lds_ids[t - 1];
            bool e = (id != PADID) && (punct[id] != 0) &&
                     !((t > 0) && (abbr[prev] != 0));     // col 0 forced false
            ew |= ((unsigned)e) << i;
        }
        e_words[w]  = ew;
        is_words[w] = 0u;
        if (localmax >= 0) atomicMax(&s_last_real, localmax);
    }
    __syncthreads();

    // ---- lane 0: boundary chain + metadata emit ----
    if (tid == 0) {
        const int last_real = s_last_real;
        int* wsrow = ws + (size_t)b * WS_ROW;
        int nb = 0;
        if (last_real >= 0) {
            int lb = -1;
            while (lb != last_real) {
                // first base_end bit at position >= lb+1
                int from = lb + 1;
                int wi = from >> 5;
                unsigned x = e_words[wi] & (0xFFFFFFFFu << (from & 31));
                int nxt = 0x7fffffff;
                while (true) {
                    if (x) { nxt = (wi << 5) + __builtin_ctz(x); break; }
                    if (++wi >= NW) break;
                    x = e_words[wi];
                }
                int next = lb + MAXP;                 // forced fire at gap 32
                if (nxt < next)       next = nxt;     // punct fire
                if (last_real < next) next = last_real; // at_last fire / clamp
                is_words[next >> 5] |= (1u << (next & 31));
                wsrow[WS_BND + nb] = next;
                ++nb;
                lb = next;
            }
        }
        wsrow[WS_META + 0] = nb;
        wsrow[WS_META + 1] = last_real;
        // exclusive prefix of phrase counts per word + is_end words
        unsigned acc = 0u;
        for (int i = 0; i < NW; ++i) {
            unsigned iw = is_words[i];
            wsrow[WS_WORDS  + i] = (int)iw;
            wsrow[WS_PREFIX + i] = (int)acc;
            acc += (unsigned)__popc(iw);
        }
    }
}

// ---------------------------------------------------------------------------
// K3: emit. One thread per (b, t):
//   phrase_id everywhere; mask/token_idx for real tokens; end_pos for t < nb.
// All metadata reads hit the ~2 MB workspace (L2-resident on 192 MB L2).
// ---------------------------------------------------------------------------
__global__ void opb_emit_kernel(const int* __restrict__ ws,
                                float* __restrict__ out) {
    const int g = blockIdx.x * blockDim.x + threadIdx.x;   // 0 .. B*T-1
    const int b = g >> 13;
    const int t = g & (TT - 1);
    const int* wsrow = ws + (size_t)b * WS_ROW;

    const int wi = t >> 5, bi = t & 31;
    const unsigned word = (unsigned)wsrow[WS_WORDS + wi];
    const int pid = wsrow[WS_PREFIX + wi] + __popc(word & ((1u << bi) - 1u));

    out[OUT_PID + (size_t)g] = (float)pid;                 // phrase_id

    const int nb        = wsrow[WS_META + 0];
    const int last_real = wsrow[WS_META + 1];

    if (t <= last_real) {                                  // real token
        int start = (pid == 0) ? 0 : (wsrow[WS_BND + pid - 1] + 1);
        int slot  = t - start;
        if (slot >= 0 && slot < MAXP) {                    // always true by construction
            size_t base = ((size_t)b * TT + (size_t)pid) * MAXP + (size_t)slot;
            out[OUT_MASK + base] = 1.0f;                   // phrase_mask
            out[OUT_TIDX + base] = (float)t;               // phrase_token_idx
        }
    }
    if (t < nb) {                                          // phrase_end_pos
        out[OUT_ENDP + (size_t)b * TT + (size_t)t] = (float)wsrow[WS_BND + t];
    }
}

// ---------------------------------------------------------------------------
extern "C" void kernel_launch(void* const* d_in, const int* in_sizes, int n_in,
                              void* d_out, int out_size, void* d_ws, size_t ws_size,
                              hipStream_t stream) {
    const int* ids   = (const int*)d_in[0];   // [B,T] int32
    const int* punct = (const int*)d_in[1];   // [VOCAB] (bool -> int)
    const int* abbr  = (const int*)d_in[2];   // [VOCAB] (bool -> int)
    float* out = (float*)d_out;
    int*   ws  = (int*)d_ws;

    (void)in_sizes; (void)n_in; (void)out_size; (void)ws_size;

    // K1: background fill (HBM write-bandwidth bound, b128 stores)
    opb_fill_kernel<<<dim3(4096), dim3(256), 0, stream>>>(out);

    // K2: per-row scan (async-to-LDS staging, serial chain per row)
    opb_scan_kernel<<<dim3(BB), dim3(256), 0, stream>>>(ids, punct, abbr, ws);

    // K3: emit all outputs
    opb_emit_kernel<<<dim3((BB * TT) / 256), dim3(256), 0, stream>>>(ws, out);
}